// DynamicInnerLoop_47528108097598
// MI455X (gfx1250) — compile-verified
//
#include <hip/hip_runtime.h>
#include <hip/hip_bf16.h>
#include <math.h>

typedef __attribute__((ext_vector_type(2))) float v2f;
typedef __attribute__((ext_vector_type(8))) float v8f;

#define LD   200064   // row stride of sp_w1 / st_w1 (N + 64)
#define NPAR 200000   // model parameter count

// ---------------------------------------------------------------------------
// Kernel 1: partial dot products via V_WMMA_F32_16X16X4_F32.
// D = A x B + C with A = 16x4 weight tile, B = 4x16 where column 0 = p0,
// column 1 = g. Columns 2-15 of B carry garbage (duplicated g data) — each
// D column depends only on its own B column, and we never read columns 2-15,
// so no masking/zeroing is needed in the hot loop. Accumulator column 0 ->
// u = W@p0 partial, column 1 -> v = W@g partial. Grid: (k-chunks, 16 row
// tiles); tile<8 -> sp_w1 rows, tile>=8 -> st_w1 rows. 4 waves per block.
// ---------------------------------------------------------------------------
__global__ __launch_bounds__(128) void k_matvec_partials(
    const float* __restrict__ sp_w1, const float* __restrict__ st_w1,
    const float* __restrict__ p0, const float* __restrict__ g,
    float* __restrict__ partials, int ksteps, int cw_total) {
  const int tile = blockIdx.y;                    // 0..15
  const float* W = (tile < 8) ? sp_w1 : st_w1;
  const int m0   = (tile & 7) * 16;
  const int wave = threadIdx.x >> 5;
  const int lane = threadIdx.x & 31;
  const int half = lane >> 4;                     // K-half within a WMMA step
  const int l16  = lane & 15;                     // A: row index; B/C: column

  const size_t k0 = (size_t)blockIdx.x * (size_t)(ksteps * 16)
                  + (size_t)wave * (size_t)(ksteps * 4);

  // A operand: lane holds W[m0+l16, k + half*2 .. +1] (ISA 32-bit A 16x4 layout)
  const float* wp = W + (size_t)(m0 + l16) * LD + k0 + (size_t)(half * 2);
  // B operand: column 0 lanes (l16==0) stream p0, everyone else streams g.
  // Lanes 2-15 duplicate lane 1's address (coalesced) and only feed discarded
  // accumulator columns.
  const float* bp = ((l16 == 0) ? p0 : g) + k0 + (size_t)(half * 2);

  v8f c = {0.f, 0.f, 0.f, 0.f, 0.f, 0.f, 0.f, 0.f};
  for (int s = 0; s < ksteps; ++s) {
    v2f a = *(const v2f*)wp;
    v2f b = *(const v2f*)bp;
    c = __builtin_amdgcn_wmma_f32_16x16x4_f32(false, a, false, b,
                                              (short)0, c, false, false);
    wp += 4;
    bp += 4;
  }

  // C/D layout: lane = N column, VGPR r = M row r (lanes 0-15) or 8+r (16-31).
  // Column 0 (lanes 0,16) = u partials, column 1 (lanes 1,17) = v partials.
  const int cw = blockIdx.x * 4 + wave;
  float* base = partials + ((size_t)tile * cw_total + cw) * 32;
  if (l16 < 2) *(v8f*)(base + l16 * 16 + half * 8) = c;
}

// ---------------------------------------------------------------------------
// Kernel 2: single workgroup. Deterministic reduction of partials, context
// encoder, enc-tail of w1, then the 10-step scalar recursion.
// ---------------------------------------------------------------------------
__global__ __launch_bounds__(256) void k_sequential(
    const float* __restrict__ partials, int cw_total,
    const float* __restrict__ context, const float* __restrict__ rand_vals,
    const float* __restrict__ sp_w1, const float* __restrict__ sp_b1,
    const float* __restrict__ sp_w2, const float* __restrict__ sp_b2,
    const float* __restrict__ sp_w3, const float* __restrict__ sp_b3,
    const float* __restrict__ st_w1, const float* __restrict__ st_b1,
    const float* __restrict__ st_w2, const float* __restrict__ st_b2,
    const float* __restrict__ st_w3, const float* __restrict__ st_b3,
    const float* __restrict__ ce_w1, const float* __restrict__ ce_b1,
    const float* __restrict__ ce_w2, const float* __restrict__ ce_b2,
    float* __restrict__ result, float* __restrict__ d_out) {
  __shared__ float u_sp[128], v_sp[128], u_st[128], v_st[128];
  __shared__ float enc[64], tmp64[64];
  __shared__ float a_sp[128], a_st[128];
  __shared__ float h1[128], h2[64], red[64];
  __shared__ float sS[1], sStep[1];
  const int t = threadIdx.x;

  // --- reduce partials: 512 outputs (16 tiles x {u[16], v[16]}) ---
  for (int o = t; o < 512; o += 256) {
    const int tile = o >> 5, idx = o & 31;
    const float* p = partials + (size_t)tile * cw_total * 32 + idx;
    float s = 0.f;
    for (int cw = 0; cw < cw_total; ++cw) s += p[(size_t)cw * 32];
    const int m = (tile & 7) * 16 + (idx & 15);
    if (tile < 8) { if (idx < 16) u_sp[m] = s; else v_sp[m] = s; }
    else          { if (idx < 16) u_st[m] = s; else v_st[m] = s; }
  }
  __syncthreads();

  // --- context encoder: enc = ce_w2 @ relu(ce_w1 @ ctx + b1) + b2 ---
  if (t < 64) {
    float s = ce_b1[t];
    for (int k = 0; k < 100; ++k) s += ce_w1[t * 100 + k] * context[k];
    tmp64[t] = fmaxf(s, 0.f);
  }
  __syncthreads();
  if (t < 64) {
    float s = ce_b2[t];
    for (int k = 0; k < 64; ++k) s += ce_w2[t * 64 + k] * tmp64[k];
    enc[t] = s;
  }
  __syncthreads();

  // --- a = u + w1[:, N:] @ enc + b1  (the S-independent part of layer 1) ---
  if (t < 128) {
    float s = sp_b1[t] + u_sp[t];
    const float* wr = sp_w1 + (size_t)t * LD + NPAR;
    for (int k = 0; k < 64; ++k) s += wr[k] * enc[k];
    a_sp[t] = s;
    float s2 = st_b1[t] + u_st[t];
    const float* wr2 = st_w1 + (size_t)t * LD + NPAR;
    for (int k = 0; k < 64; ++k) s2 += wr2[k] * enc[k];
    a_st[t] = s2;
  }
  if (t == 0) sS[0] = 0.f;
  __syncthreads();

  // --- 10-step recursion: layer1 output = relu(a - S * v) ---
  int count = 0;
  bool active = true;
  for (int step = 0; step < 10; ++step) {
    const float S = sS[0];
    // step-size head
    if (t < 128) h1[t] = fmaxf(a_sp[t] - S * v_sp[t], 0.f);
    __syncthreads();
    if (t < 64) {
      float s = sp_b2[t];
      for (int k = 0; k < 128; ++k) s += sp_w2[t * 128 + k] * h1[k];
      h2[t] = fmaxf(s, 0.f);
    }
    __syncthreads();
    if (t < 64) red[t] = sp_w3[t] * h2[t];
    __syncthreads();
    if (t == 0) {
      float s = sp_b3[0];
      for (int k = 0; k < 64; ++k) s += red[k];
      sStep[0] = 0.1f / (1.f + expf(-s));      // sigmoid * 0.1
    }
    __syncthreads();
    // stop head (same pre-update S)
    if (t < 128) h1[t] = fmaxf(a_st[t] - S * v_st[t], 0.f);
    __syncthreads();
    if (t < 64) {
      float s = st_b2[t];
      for (int k = 0; k < 128; ++k) s += st_w2[t * 128 + k] * h1[k];
      h2[t] = fmaxf(s, 0.f);
    }
    __syncthreads();
    if (t < 64) red[t] = st_w3[t] * h2[t];
    __syncthreads();
    if (t == 0) {
      float s = st_b3[0];
      for (int k = 0; k < 64; ++k) s += red[k];
      const float stop_prob = 1.f / (1.f + expf(-s));
      if (active) { sS[0] = S + sStep[0]; count++; }   // update before stop check
      if (step >= 1 && rand_vals[step] < stop_prob) active = false;
    }
    __syncthreads();
  }

  if (t == 0) {
    result[0] = sS[0];
    result[1] = (float)count;
    d_out[NPAR] = (float)count;   // second tuple output
  }
}

// ---------------------------------------------------------------------------
// Kernel 3: out[i] = p0[i] - S_final * g[i]
// ---------------------------------------------------------------------------
__global__ __launch_bounds__(256) void k_finalize(
    const float* __restrict__ p0, const float* __restrict__ g,
    const float* __restrict__ result, float* __restrict__ out, int n) {
  const int i = blockIdx.x * blockDim.x + threadIdx.x;
  if (i < n) out[i] = p0[i] - result[0] * g[i];
}

// ---------------------------------------------------------------------------
extern "C" void kernel_launch(void* const* d_in, const int* in_sizes, int n_in,
                              void* d_out, int out_size, void* d_ws, size_t ws_size,
                              hipStream_t stream) {
  const float* p0        = (const float*)d_in[0];
  const float* g         = (const float*)d_in[1];
  const float* context   = (const float*)d_in[2];
  // d_in[3] = loss_history (unused by the reference)
  const float* rand_vals = (const float*)d_in[4];
  const float* sp_w1 = (const float*)d_in[5];
  const float* sp_b1 = (const float*)d_in[6];
  const float* sp_w2 = (const float*)d_in[7];
  const float* sp_b2 = (const float*)d_in[8];
  const float* sp_w3 = (const float*)d_in[9];
  const float* sp_b3 = (const float*)d_in[10];
  const float* st_w1 = (const float*)d_in[11];
  const float* st_b1 = (const float*)d_in[12];
  const float* st_w2 = (const float*)d_in[13];
  const float* st_b2 = (const float*)d_in[14];
  const float* st_w3 = (const float*)d_in[15];
  const float* st_b3 = (const float*)d_in[16];
  const float* ce_w1 = (const float*)d_in[17];
  const float* ce_b1 = (const float*)d_in[18];
  const float* ce_w2 = (const float*)d_in[19];
  const float* ce_b2 = (const float*)d_in[20];
  float* out = (float*)d_out;

  // Pick the largest k-chunk count whose partial buffer fits in d_ws.
  // Requirement: NPAR / (chunks*16) integral  (4 waves x 4 K per WMMA step).
  const int cand[6] = {50, 25, 10, 5, 2, 1};
  int chunks = 1;
  for (int i = 0; i < 6; ++i) {
    const size_t need = (size_t)16 * cand[i] * 4 * 32 * sizeof(float) + 64;
    if (need <= ws_size) { chunks = cand[i]; break; }
  }
  const int cw_total = chunks * 4;                 // partial vectors per tile
  const int ksteps   = NPAR / (chunks * 16);       // WMMA K-steps per wave

  float* partials = (float*)d_ws;
  float* result   = partials + (size_t)16 * cw_total * 32;

  dim3 grid1(chunks, 16);
  k_matvec_partials<<<grid1, dim3(128), 0, stream>>>(sp_w1, st_w1, p0, g,
                                                     partials, ksteps, cw_total);

  k_sequential<<<1, 256, 0, stream>>>(partials, cw_total, context, rand_vals,
                                      sp_w1, sp_b1, sp_w2, sp_b2, sp_w3, sp_b3,
                                      st_w1, st_b1, st_w2, st_b2, st_w3, st_b3,
                                      ce_w1, ce_b1, ce_w2, ce_b2,
                                      result, out);

  k_finalize<<<dim3((NPAR + 255) / 256), dim3(256), 0, stream>>>(p0, g, result,
                                                                 out, NPAR);
}